// Model_30605936951775
// MI455X (gfx1250) — compile-verified
//
#include <hip/hip_runtime.h>
#include <hip/hip_bf16.h>
#include <math.h>

// ---------------- problem constants ----------------
#define HH 256
#define WW 256
#define PIX (HH * WW)
#define NFACES 640
#define NVERTS 322
#define TS 4
#define NTILES (NFACES / 16)   // 40 face tiles of 16
#define EYE_Z 2.732f
#define FOCALC 3.7320508075688776f   // 1/tan(15 deg)
#define BIGF 10000000000.0f
#define EPSF 1e-08f

typedef __attribute__((ext_vector_type(2))) float v2f;
typedef __attribute__((ext_vector_type(8))) float v8f;

// ws layout (floats):
//   [0 .. 6399]   face coefficient table: 10 arrays of NFACES
//                 k*NFACES+f : k = A0,B0,C0,A1,B1,C1,Az,Bz,Cz,valid
//   [6400..6911]  512 per-block partial loss sums
#define WS_FACE 0
#define WS_BSUM 6400
#define NBLOCKS 512

// ---------------------------------------------------------------------------
// Kernel 1: per-face projection + affine edge/depth coefficients
// w0(px,py) = A0*px + B0*py + C0 ;  w1 = A1*px + B1*py + C1
// z (px,py) = Az*px + Bz*py + Cz ;  w2 = 1 - w0 - w1
// ---------------------------------------------------------------------------
__global__ void face_precompute(const float* __restrict__ verts,
                                const int*   __restrict__ faces,
                                const float* __restrict__ angle,
                                float* __restrict__ fd)
{
    int fi = blockIdx.x * blockDim.x + threadIdx.x;
    if (fi >= NFACES) return;

    float th = angle[0] * 6.283185307179586f;
    float c = cosf(th), s = sinf(th);

    float PXv[3], PYv[3], FD[3];
#pragma unroll
    for (int k = 0; k < 3; ++k) {
        int vi = faces[fi * 3 + k];
        float x = verts[vi * 3 + 0];
        float y = verts[vi * 3 + 1];
        float z = verts[vi * 3 + 2];
        float xr = c * x + s * z;
        float zr = -s * x + c * z;
        float depth = EYE_Z - zr;
        PXv[k] = FOCALC * xr / depth;
        PYv[k] = FOCALC * y / depth;
        FD[k]  = depth;
    }
    float ax = PXv[0], bx = PXv[1], cx = PXv[2];
    float ay = PYv[0], by = PYv[1], cy = PYv[2];

    float area   = (bx - ax) * (cy - ay) - (by - ay) * (cx - ax);
    float validf = (fabsf(area) > EPSF) ? 1.0f : 0.0f;
    float denom  = (validf != 0.0f) ? area : 1.0f;
    float inv    = 1.0f / denom;

    // edge(bx,by,cx,cy): w0 = ((cx-bx)*(py-by) - (cy-by)*(px-bx)) / denom
    float A0 = -(cy - by) * inv;
    float B0 =  (cx - bx) * inv;
    float C0 = ((cy - by) * bx - (cx - bx) * by) * inv;
    // edge(cx,cy,ax,ay): w1
    float A1 = -(ay - cy) * inv;
    float B1 =  (ax - cx) * inv;
    float C1 = ((ay - cy) * cx - (ax - cx) * cy) * inv;

    // z = w0*fd0 + w1*fd1 + (1-w0-w1)*fd2
    float df0 = FD[0] - FD[2];
    float df1 = FD[1] - FD[2];
    float Az = A0 * df0 + A1 * df1;
    float Bz = B0 * df0 + B1 * df1;
    float Cz = C0 * df0 + C1 * df1 + FD[2];

    fd[0 * NFACES + fi] = A0;
    fd[1 * NFACES + fi] = B0;
    fd[2 * NFACES + fi] = C0;
    fd[3 * NFACES + fi] = A1;
    fd[4 * NFACES + fi] = B1;
    fd[5 * NFACES + fi] = C1;
    fd[6 * NFACES + fi] = Az;
    fd[7 * NFACES + fi] = Bz;
    fd[8 * NFACES + fi] = Cz;
    fd[9 * NFACES + fi] = validf;
}

// ---------------------------------------------------------------------------
// Kernel 2: WMMA-tiled rasterizer. Each wave32 owns 16 pixels (one image row
// segment). Per 16-face tile: 3x v_wmma_f32_16x16x4_f32 evaluate (w0,w1,z)
// for the 16x16 pixel-face tile; branchless elementwise z-buffer argmin on
// the accumulator layout; cross-lane min-reduce; trilinear sample + SSE.
// ---------------------------------------------------------------------------
__global__ __launch_bounds__(256) void raster_kernel(
    const float* __restrict__ faceData,
    const float* __restrict__ textures,   // raw (pre-tanh), (F,4,4,4,3)
    const float* __restrict__ image_ref,  // (3,H,W)
    float* __restrict__ blockSums)
{
    __shared__ float ldsF[10 * NFACES];
    __shared__ float red[256];

    // stage coefficient table in LDS (25.6 KB of the 320 KB WGP LDS)
    for (int i = threadIdx.x; i < 10 * NFACES; i += 256)
        ldsF[i] = faceData[i];
    __syncthreads();

    const int lane     = threadIdx.x & 31;
    const int halfLane = lane & 15;
    const bool hi      = (lane >= 16);
    const int waveId   = blockIdx.x * 8 + (threadIdx.x >> 5);
    const int pixBase  = waveId * 16;                 // 16 consecutive pixels
    const int iy       = pixBase >> 8;                // /W (W=256)
    const int ix0      = pixBase & 255;
    const float pyv = 1.0f - (iy + 0.5f) * (2.0f / HH);
    const float pxv = ((float)(ix0 + halfLane) + 0.5f) * (2.0f / WW) - 1.0f;

    // A matrix 16x4 (M=pixel row, K = [px, py, 1, 0]):
    // lanes 0-15 hold K=0,1 ; lanes 16-31 hold K=2,3  (ISA 7.12.2)
    v2f Aop;
    Aop.x = hi ? 1.0f : pxv;
    Aop.y = hi ? 0.0f : pyv;

    // Branch-free B operand construction: lanes 16-31 need the C-coefficient
    // row in .x (address select, single ds_load) and zero in .y (value select).
    const int xsel = hi ? (2 * NFACES) : 0;   // +2 arrays: A->C row
    const float ymul = hi ? 0.0f : 1.0f;      // zero out K=3 row

    float bestz[8], bw0[8], bw1[8];
    int   bf[8];
#pragma unroll
    for (int r = 0; r < 8; ++r) {
        bestz[r] = BIGF; bw0[r] = 0.0f; bw1[r] = 0.0f; bf[r] = 0x7FFFFFFF;
    }

    v8f czero = {};

    for (int t = 0; t < NTILES; ++t) {
        const int face = t * 16 + halfLane;           // this lane's column N
        // B matrix 4x16 (K rows = [A,B,C,0] coeffs, N = face column):
        // lanes 0-15 hold K=0,1 ; lanes 16-31 hold K=2,3
        v2f B0op, B1op, Bzop;
        B0op.x = ldsF[0 * NFACES + face + xsel];
        B0op.y = ldsF[1 * NFACES + face] * ymul;
        B1op.x = ldsF[3 * NFACES + face + xsel];
        B1op.y = ldsF[4 * NFACES + face] * ymul;
        Bzop.x = ldsF[6 * NFACES + face + xsel];
        Bzop.y = ldsF[7 * NFACES + face] * ymul;
        const float fvalid = ldsF[9 * NFACES + face];

        v8f w0m = __builtin_amdgcn_wmma_f32_16x16x4_f32(
            false, Aop, false, B0op, (short)0, czero, false, false);
        v8f w1m = __builtin_amdgcn_wmma_f32_16x16x4_f32(
            false, Aop, false, B1op, (short)0, czero, false, false);
        v8f zm = __builtin_amdgcn_wmma_f32_16x16x4_f32(
            false, Aop, false, Bzop, (short)0, czero, false, false);

        // branchless z-buffer update on the accumulator layout:
        // VGPR slot r = pixel row (halfBase + r), lane = face column.
#pragma unroll
        for (int r = 0; r < 8; ++r) {
            float w0 = w0m[r];
            float w1 = w1m[r];
            float zz = zm[r];
            float w2 = 1.0f - w0 - w1;
            bool inside = (w0 >= 0.0f) & (w1 >= 0.0f) & (w2 >= 0.0f) &
                          (fvalid != 0.0f);
            float zb = (inside && (zz > EPSF)) ? zz : BIGF;
            bool take = zb < bestz[r];    // strict <  => first face index wins
            bestz[r] = take ? zb   : bestz[r];
            bf[r]    = take ? face : bf[r];
            bw0[r]   = take ? w0   : bw0[r];
            bw1[r]   = take ? w1   : bw1[r];
        }
    }

    // cross-lane min+argmin within each 16-lane half (ties -> smaller face)
#pragma unroll
    for (int r = 0; r < 8; ++r) {
#pragma unroll
        for (int m = 1; m < 16; m <<= 1) {
            float oz  = __shfl_xor(bestz[r], m, 32);
            int   of  = __shfl_xor(bf[r],    m, 32);
            float ow0 = __shfl_xor(bw0[r],   m, 32);
            float ow1 = __shfl_xor(bw1[r],   m, 32);
            bool take = (oz < bestz[r]) || (oz == bestz[r] && of < bf[r]);
            bestz[r] = take ? oz  : bestz[r];
            bf[r]    = take ? of  : bf[r];
            bw0[r]   = take ? ow0 : bw0[r];
            bw1[r]   = take ? ow1 : bw1[r];
        }
    }

    // trilinear texture sample + per-pixel SSE: lanes 0-7 handle pixel rows
    // 0-7, lanes 16-23 handle rows 8-15 (all lanes of a half hold the full
    // reduced result, so lane k selects row k).
    float loss = 0.0f;
    if (halfLane < 8) {
        float sz = bestz[0], sw0 = bw0[0], sw1 = bw1[0];
        int sf = bf[0];
#pragma unroll
        for (int r = 1; r < 8; ++r) {
            bool sel = (halfLane == r);
            sz  = sel ? bestz[r] : sz;
            sw0 = sel ? bw0[r]   : sw0;
            sw1 = sel ? bw1[r]   : sw1;
            sf  = sel ? bf[r]    : sf;
        }
        const int rowInWave = (hi ? 8 : 0) + halfLane;
        const int pix = pixBase + rowInWave;

        const bool hit = sz < (0.5f * BIGF);
        const int fi = hit ? sf : 0;

        float pw0 = fminf(fmaxf(sw0, 0.0f), 1.0f);
        float pw1 = fminf(fmaxf(sw1, 0.0f), 1.0f);
        float pw2 = fminf(fmaxf(1.0f - sw0 - sw1, 0.0f), 1.0f);
        float u0 = pw0 * (TS - 1), u1 = pw1 * (TS - 1), u2 = pw2 * (TS - 1);
        float f0 = fminf(fmaxf(floorf(u0), 0.0f), (float)(TS - 2));
        float f1 = fminf(fmaxf(floorf(u1), 0.0f), (float)(TS - 2));
        float f2 = fminf(fmaxf(floorf(u2), 0.0f), (float)(TS - 2));
        int i00 = (int)f0, i01 = (int)f1, i02 = (int)f2;
        float fr0 = u0 - f0, fr1 = u1 - f1, fr2 = u2 - f2;

        float col0 = 0.0f, col1 = 0.0f, col2 = 0.0f;
        const float* texf = textures + (size_t)fi * (TS * TS * TS * 3);
#pragma unroll
        for (int a = 0; a < 2; ++a) {
            float wa = a ? fr0 : (1.0f - fr0);
#pragma unroll
            for (int b = 0; b < 2; ++b) {
                float wb = b ? fr1 : (1.0f - fr1);
#pragma unroll
                for (int cc = 0; cc < 2; ++cc) {
                    float wc = (cc ? fr2 : (1.0f - fr2)) * wa * wb;
                    int idxf = (i00 + a) * (TS * TS) + (i01 + b) * TS + (i02 + cc);
                    const float* tp = texf + idxf * 3;
                    col0 += wc * tanhf(tp[0]);
                    col1 += wc * tanhf(tp[1]);
                    col2 += wc * tanhf(tp[2]);
                }
            }
        }
        float c0v = hit ? col0 : 0.0f;
        float c1v = hit ? col1 : 0.0f;
        float c2v = hit ? col2 : 0.0f;
        float d0 = c0v - image_ref[0 * PIX + pix];
        float d1 = c1v - image_ref[1 * PIX + pix];
        float d2 = c2v - image_ref[2 * PIX + pix];
        loss = d0 * d0 + d1 * d1 + d2 * d2;
    }

    // deterministic block reduction
    red[threadIdx.x] = loss;
    __syncthreads();
#pragma unroll
    for (int s2 = 128; s2 > 0; s2 >>= 1) {
        if ((int)threadIdx.x < s2) red[threadIdx.x] += red[threadIdx.x + s2];
        __syncthreads();
    }
    if (threadIdx.x == 0) blockSums[blockIdx.x] = red[0];
}

// ---------------------------------------------------------------------------
// Kernel 3: deterministic final reduction of 512 block partials -> scalar
// ---------------------------------------------------------------------------
__global__ __launch_bounds__(256) void final_reduce(const float* __restrict__ bs,
                                                    float* __restrict__ out)
{
    __shared__ float red[256];
    int tid = threadIdx.x;
    red[tid] = bs[tid] + bs[tid + 256];
    __syncthreads();
#pragma unroll
    for (int s2 = 128; s2 > 0; s2 >>= 1) {
        if (tid < s2) red[tid] += red[tid + s2];
        __syncthreads();
    }
    if (tid == 0) out[0] = red[0];
}

// ---------------------------------------------------------------------------
extern "C" void kernel_launch(void* const* d_in, const int* in_sizes, int n_in,
                              void* d_out, int out_size, void* d_ws, size_t ws_size,
                              hipStream_t stream)
{
    const float* verts     = (const float*)d_in[0];  // (1,322,3) f32
    const int*   faces     = (const int*)  d_in[1];  // (1,640,3) i32
    const float* textures  = (const float*)d_in[2];  // (1,640,4,4,4,3) f32
    const float* image_ref = (const float*)d_in[3];  // (1,3,256,256) f32
    const float* angle     = (const float*)d_in[4];  // (1,) f32

    float* ws        = (float*)d_ws;
    float* faceData  = ws + WS_FACE;
    float* blockSums = ws + WS_BSUM;

    face_precompute<<<(NFACES + 255) / 256, 256, 0, stream>>>(
        verts, faces, angle, faceData);
    raster_kernel<<<NBLOCKS, 256, 0, stream>>>(
        faceData, textures, image_ref, blockSums);
    final_reduce<<<1, 256, 0, stream>>>(blockSums, (float*)d_out);
}